// MEWISPool_82420422410981
// MI455X (gfx1250) — compile-verified
//
#include <hip/hip_runtime.h>
#include <hip/hip_bf16.h>
#include <math.h>

#define Nn   2048
#define Dd   512
#define HIDD 512
#define NG   16

typedef __attribute__((ext_vector_type(2))) float v2f;
typedef __attribute__((ext_vector_type(8))) float v8f;

// ---------------------------------------------------------------------------
// generic block reduction helpers (blockDim.x threads, power of two)
// ---------------------------------------------------------------------------
__device__ __forceinline__ float block_reduce_sum(float val, float* red) {
    int tid = threadIdx.x;
    red[tid] = val; __syncthreads();
    for (int off = blockDim.x >> 1; off > 0; off >>= 1) {
        if (tid < off) red[tid] += red[tid + off];
        __syncthreads();
    }
    float r = red[0]; __syncthreads();
    return r;
}
__device__ __forceinline__ float block_reduce_max(float val, float* red) {
    int tid = threadIdx.x;
    red[tid] = val; __syncthreads();
    for (int off = blockDim.x >> 1; off > 0; off >>= 1) {
        if (tid < off) red[tid] = fmaxf(red[tid], red[tid + off]);
        __syncthreads();
    }
    float r = red[0]; __syncthreads();
    return r;
}

// ---------------------------------------------------------------------------
// CDNA5 async global->LDS 128-bit copy (ASYNCcnt tracked, no VGPR round trip)
// LDS operand = wave-relative LDS byte offset = low 32 bits of the flat addr.
// ---------------------------------------------------------------------------
__device__ __forceinline__ void async_b128(void* lds, const void* gaddr) {
    asm volatile("global_load_async_to_lds_b128 %0, %1, off"
                 :: "v"((unsigned)(size_t)lds), "v"(gaddr)
                 : "memory");
}
__device__ __forceinline__ void wait_async0() {
    asm volatile("s_wait_asynccnt 0" ::: "memory");
}

// ---------------------------------------------------------------------------
// fp32 WMMA GEMM:  C = act( A[MxK] @ B[KxN] (+bias) (+addend) )
//   block = 256 threads = 8 waves; macro tile 64(M) x 32(N); K tile 32
//   each wave owns a 16x16 C tile via V_WMMA_F32_16X16X4_F32
//   tiles staged with global_load_async_to_lds_b128, double-buffered
// ---------------------------------------------------------------------------
#define BM 64
#define BN 32
#define KT 32
#define APAD 4   // rows padded to 36 floats -> 144B row stride (16B aligned)

__device__ __forceinline__ void stage_tiles(const float* __restrict__ Ag,
                                            const float* __restrict__ Bg,
                                            int K, int Ncols, int blockM, int blockN,
                                            int k0, int tid, int squareB,
                                            float (*As)[KT + APAD],
                                            float (*Bs)[BN + APAD]) {
    // A tile: BM x KT = 2048 floats = 512 x 16B chunks, 2 per thread (async)
    #pragma unroll
    for (int e = 0; e < 2; ++e) {
        int ch = tid + e * 256;
        int r  = ch >> 3;
        int cc = (ch & 7) << 2;
        async_b128(&As[r][cc], Ag + (size_t)(blockM + r) * K + (k0 + cc));
    }
    // B tile: KT x BN = 1024 floats = 256 x 16B chunks, 1 per thread
    int r  = tid >> 3;
    int cc = (tid & 7) << 2;
    if (squareB) {
        float4 bv = *(const float4*)(Bg + (size_t)(k0 + r) * Ncols + (blockN + cc));
        bv.x *= bv.x; bv.y *= bv.y; bv.z *= bv.z; bv.w *= bv.w;
        *(float4*)&Bs[r][cc] = bv;
    } else {
        async_b128(&Bs[r][cc], Bg + (size_t)(k0 + r) * Ncols + (blockN + cc));
    }
}

__device__ __forceinline__ v8f tile_mma(const float (*As)[KT + APAD],
                                        const float (*Bs)[BN + APAD],
                                        int arow, int bcol, int hi, v8f c) {
    #pragma unroll
    for (int kk = 0; kk < KT; kk += 4) {
        int kp = kk + hi * 2;
        v2f a, b;
        a.x = As[arow][kp];
        a.y = As[arow][kp + 1];
        b.x = Bs[kp][bcol];
        b.y = Bs[kp + 1][bcol];
        c = __builtin_amdgcn_wmma_f32_16x16x4_f32(
                false, a, false, b, (short)0, c, false, false);
    }
    return c;
}

__global__ __launch_bounds__(256)
void gemm_f32_wmma(const float* __restrict__ Ag, const float* __restrict__ Bg,
                   const float* __restrict__ bias, const float* __restrict__ addend,
                   float* __restrict__ Cg, int M, int Ncols, int K,
                   int do_relu, int squareB)
{
    __shared__ __align__(16) float As[2][BM][KT + APAD];
    __shared__ __align__(16) float Bs[2][KT][BN + APAD];

    const int tid   = threadIdx.x;
    const int wid   = tid >> 5;       // wave id 0..7 (wave32)
    const int lane  = tid & 31;
    const int waveM = wid >> 1;       // 0..3
    const int waveN = wid & 1;        // 0..1
    const int row16 = lane & 15;
    const int hi    = lane >> 4;      // 0/1: K-pair selector (A,B), M+8 selector (C)

    const int blockM = blockIdx.y * BM;
    const int blockN = blockIdx.x * BN;

    v8f c;
    #pragma unroll
    for (int r = 0; r < 8; ++r) c[r] = 0.0f;

    const int arow = waveM * 16 + row16;
    const int bcol = waveN * 16 + row16;
    const int nst  = K / KT;

    // prologue: stage tile 0
    stage_tiles(Ag, Bg, K, Ncols, blockM, blockN, 0, tid, squareB, As[0], Bs[0]);
    wait_async0();
    __syncthreads();

    for (int s = 0; s < nst; ++s) {
        int cur = s & 1;
        if (s + 1 < nst)
            stage_tiles(Ag, Bg, K, Ncols, blockM, blockN, (s + 1) * KT, tid,
                        squareB, As[cur ^ 1], Bs[cur ^ 1]);
        c = tile_mma(As[cur], Bs[cur], arow, bcol, hi, c);
        wait_async0();          // next tile landed
        __syncthreads();        // everyone done reading current tile
    }

    // epilogue: C tile 16x16, VGPR r -> M = r + 8*hi
    const int colg = blockN + waveN * 16 + row16;
    #pragma unroll
    for (int r = 0; r < 8; ++r) {
        int rowg = blockM + waveM * 16 + hi * 8 + r;
        float val = c[r];
        if (addend) val += addend[(size_t)rowg * Ncols + colg];
        if (bias)   val += bias[colg];
        if (do_relu) val = fmaxf(val, 0.0f);
        Cg[(size_t)rowg * Ncols + colg] = val;
    }
}

// ---------------------------------------------------------------------------
// CSR extraction from dense 0/1 adjacency
// ---------------------------------------------------------------------------
__global__ void k_deg(const float* __restrict__ A, int* __restrict__ deg) {
    __shared__ int red[256];
    int row = blockIdx.x, tid = threadIdx.x;
    const float* Ar = A + (size_t)row * Nn;
    int cnt = 0;
    for (int j = tid; j < Nn; j += 256) cnt += (Ar[j] == 1.0f);
    red[tid] = cnt; __syncthreads();
    for (int off = 128; off > 0; off >>= 1) {
        if (tid < off) red[tid] += red[tid + off];
        __syncthreads();
    }
    if (tid == 0) deg[row] = red[0];
}

__global__ void k_scan(const int* __restrict__ deg, int* __restrict__ rowptr) {
    if (threadIdx.x == 0 && blockIdx.x == 0) {
        int a = 0;
        for (int i = 0; i < Nn; ++i) { rowptr[i] = a; a += deg[i]; }
        rowptr[Nn] = a;
    }
}

__global__ void k_fill(const float* __restrict__ A, const int* __restrict__ rowptr,
                       int* __restrict__ colidx) {
    __shared__ int sc[256];
    int row = blockIdx.x, tid = threadIdx.x;
    const float* Ar = A + (size_t)row * Nn;
    const int c0 = tid * 8;                  // 256*8 = 2048 cols
    int local = 0;
    #pragma unroll
    for (int j = 0; j < 8; ++j) local += (Ar[c0 + j] == 1.0f);
    sc[tid] = local; __syncthreads();
    for (int off = 1; off < 256; off <<= 1) {
        int t = (tid >= off) ? sc[tid - off] : 0;
        __syncthreads();
        sc[tid] += t;
        __syncthreads();
    }
    int pos = rowptr[row] + (sc[tid] - local);   // exclusive prefix -> ordered fill
    #pragma unroll
    for (int j = 0; j < 8; ++j)
        if (Ar[c0 + j] == 1.0f) colidx[pos++] = c0 + j;
}

// ---------------------------------------------------------------------------
// V row norms:  v[i] = || x^2 - 2 x*Ax + A@(x^2) ||_row
// ---------------------------------------------------------------------------
__global__ void k_vnorm(const float* __restrict__ x, const float* __restrict__ Ax,
                        const float* __restrict__ Axx, float* __restrict__ v) {
    __shared__ float red[256];
    int row = blockIdx.x, tid = threadIdx.x;
    float acc = 0.0f;
    for (int j = tid; j < Dd; j += 256) {
        size_t id = (size_t)row * Dd + j;
        float xv = x[id];
        float t = xv * xv - 2.0f * xv * Ax[id] + Axx[id];
        acc += t * t;
    }
    float s = block_reduce_sum(acc, red);
    if (tid == 0) v[row] = sqrtf(s);
}

// segment softmax -> entropy
__global__ void k_segmax(const float* __restrict__ v, const int* __restrict__ batch,
                         float* __restrict__ mseg) {
    __shared__ float red[256];
    int g = blockIdx.x, tid = threadIdx.x;
    float m = -INFINITY;
    for (int i = tid; i < Nn; i += 256) if (batch[i] == g) m = fmaxf(m, v[i]);
    float r = block_reduce_max(m, red);
    if (tid == 0) mseg[g] = r;
}
__global__ void k_segz(const float* __restrict__ v, const int* __restrict__ batch,
                       const float* __restrict__ mseg, float* __restrict__ zseg) {
    __shared__ float red[256];
    int g = blockIdx.x, tid = threadIdx.x;
    float mg = mseg[g], s = 0.0f;
    for (int i = tid; i < Nn; i += 256) if (batch[i] == g) s += expf(v[i] - mg);
    float r = block_reduce_sum(s, red);
    if (tid == 0) zseg[g] = r;
}
__global__ void k_ent(const float* __restrict__ v, const int* __restrict__ batch,
                      const float* __restrict__ mseg, const float* __restrict__ zseg,
                      float* __restrict__ ent) {
    int i = blockIdx.x * 256 + threadIdx.x;
    if (i >= Nn) return;
    int g = batch[i];
    float e = expf(v[i] - mseg[g]);
    float P = e / zseg[g];
    ent[i] = (P == 0.0f) ? 0.0f : -P * logf(P);
}

// t1 = ent + A@ent  (sparse matvec over CSR; A is 0/1)
__global__ void k_t1(const float* __restrict__ ent, const int* __restrict__ rowptr,
                     const int* __restrict__ colidx, float* __restrict__ t1) {
    int i = blockIdx.x * 256 + threadIdx.x;
    if (i >= Nn) return;
    float s = ent[i];
    for (int p = rowptr[i]; p < rowptr[i + 1]; ++p) s += ent[colidx[p]];
    t1[i] = s;
}

// layer-1 entry: h[i,o] = relu(t1[i]*w11[o] + b11[o])
__global__ void k_layer1(const float* __restrict__ t1, const float* __restrict__ w11,
                         const float* __restrict__ b11, float* __restrict__ h) {
    size_t idx = (size_t)blockIdx.x * 256 + threadIdx.x;
    if (idx >= (size_t)Nn * HIDD) return;
    int i = (int)(idx / HIDD), o = (int)(idx % HIDD);
    h[idx] = fmaxf(t1[i] * w11[o] + b11[o], 0.0f);
}

// h3 / sigmoid:  probs[i] = sigmoid(dot(g[i,:], w33) + b33)
__global__ void k_h3(const float* __restrict__ g, const float* __restrict__ w33,
                     const float* __restrict__ b33, float* __restrict__ probs) {
    __shared__ float red[256];
    int row = blockIdx.x, tid = threadIdx.x;
    float acc = 0.0f;
    for (int k = tid; k < HIDD; k += 256) acc += g[(size_t)row * HIDD + k] * w33[k];
    float s = block_reduce_sum(acc, red);
    if (tid == 0) probs[row] = 1.0f / (1.0f + expf(-(s + b33[0])));
}

// ---------------------------------------------------------------------------
// sequential greedy MEWIS selection — one 1024-thread block, LDS resident
// ---------------------------------------------------------------------------
__global__ __launch_bounds__(1024)
void k_select(const float* __restrict__ entg, const float* __restrict__ probsg,
              const int* __restrict__ rowptr, const int* __restrict__ colidx,
              float* __restrict__ out_loss, float* __restrict__ out_mask)
{
    __shared__ float e[Nn];
    __shared__ float dummy[Nn];
    __shared__ float sv[Nn];
    __shared__ float key[Nn];
    __shared__ int   idx[Nn];
    __shared__ unsigned char selA[Nn], rejA[Nn], nfl[Nn];
    __shared__ float red[1024];

    const int tid = threadIdx.x;
    for (int i = tid; i < Nn; i += 1024) {
        e[i] = entg[i];
        float p = probsg[i];
        dummy[i] = p; key[i] = p; idx[i] = i;
        selA[i] = 0; rejA[i] = 0; nfl[i] = 0;
    }
    __syncthreads();

    // gamma, dot(e,probs), probs@(A@probs)
    float g1 = 0, g2 = 0, g3 = 0;
    for (int i = tid; i < Nn; i += 1024) {
        float pi = dummy[i];
        g1 += e[i];
        g2 += e[i] * pi;
        float a = 0;
        for (int p = rowptr[i]; p < rowptr[i + 1]; ++p) a += dummy[colidx[p]];
        g3 += pi * a;
    }
    float gam = block_reduce_sum(g1, red);
    float ds  = block_reduce_sum(g2, red);
    float q0  = block_reduce_sum(g3, red);
    float thr = gam - ds + q0;                    // loss
    if (tid == 0) out_loss[0] = thr;
    __syncthreads();

    // bitonic argsort: descending probs, ties -> ascending index (stable argsort of -p)
    for (unsigned k = 2; k <= (unsigned)Nn; k <<= 1) {
        for (unsigned j = k >> 1; j > 0; j >>= 1) {
            for (int ii = tid; ii < Nn; ii += 1024) {
                unsigned p = (unsigned)ii ^ j;
                if (p > (unsigned)ii) {
                    bool up = ((ii & k) == 0);
                    float ka = key[ii], kb = key[p];
                    int ia = idx[ii], ib = idx[p];
                    bool inorder = (ka > kb) || (ka == kb && ia < ib);
                    if (up ? !inorder : inorder) {
                        key[ii] = kb; key[p] = ka;
                        idx[ii] = ib; idx[p] = ia;
                    }
                }
            }
            __syncthreads();
        }
    }

    // greedy loop
    for (int it = 0; it < Nn; ++it) {
        int node = idx[it];
        int rs = rowptr[node], re = rowptr[node + 1];
        bool has_n = (re > rs);
        bool fresh = !(selA[node] | rejA[node]);
        if (!has_n) {
            if (tid == 0) selA[node] = 1;
            __syncthreads();
            continue;
        }
        if (!fresh) { __syncthreads(); continue; }

        // mark neighbors + partial sum of e_j * dummy_j over neighborhood
        float dl = 0;
        for (int p = rs + tid; p < re; p += 1024) {
            int j = colidx[p];
            nfl[j] = 1;
            dl += e[j] * dummy[j];
        }
        __syncthreads();
        // build candidate s
        for (int i = tid; i < Nn; i += 1024) sv[i] = nfl[i] ? 0.0f : dummy[i];
        __syncthreads();
        if (tid == 0) sv[node] = 1.0f;
        __syncthreads();

        // q = s @ (A @ s) over CSR
        float ql = 0;
        for (int i = tid; i < Nn; i += 1024) {
            float si = sv[i];
            if (si != 0.0f) {
                float a = 0;
                for (int p = rowptr[i]; p < rowptr[i + 1]; ++p) a += sv[colidx[p]];
                ql += si * a;
            }
        }
        float dn = block_reduce_sum(dl, red);
        float q  = block_reduce_sum(ql, red);

        float dots = ds - dn + e[node] * (1.0f - dummy[node]);
        float l = gam - dots + q;
        bool accept = (l <= thr);

        if (accept) {
            for (int i = tid; i < Nn; i += 1024) dummy[i] = sv[i];
            if (tid == 0) selA[node] = 1;
            for (int p = rs + tid; p < re; p += 1024) rejA[colidx[p]] = 1;
            ds = dots;
        }
        // clear neighbor flags
        for (int p = rs + tid; p < re; p += 1024) nfl[colidx[p]] = 0;
        __syncthreads();
    }

    for (int i = tid; i < Nn; i += 1024) out_mask[i] = selA[i] ? 1.0f : 0.0f;
}

// ---------------------------------------------------------------------------
// final outputs
// ---------------------------------------------------------------------------
__global__ void k_hpooled(const float* __restrict__ x, const float* __restrict__ mask,
                          float* __restrict__ out) {
    size_t idx = (size_t)blockIdx.x * 256 + threadIdx.x;
    if (idx >= (size_t)Nn * Dd) return;
    int i = (int)(idx / Dd);
    out[idx] = x[idx] * mask[i];
}

__global__ void k_adjp(float* __restrict__ adj /* in: A2, out: adj_pooled */,
                       const float* __restrict__ A3, const float* __restrict__ mask) {
    size_t idx = (size_t)blockIdx.x * 256 + threadIdx.x;
    if (idx >= (size_t)Nn * Nn) return;
    int i = (int)(idx / Nn), j = (int)(idx % Nn);
    float s = adj[idx] + A3[idx];
    float cl = fminf(fmaxf(s, 0.0f), 1.0f);
    adj[idx] = (i == j) ? 0.0f : cl * mask[i] * mask[j];
}

// ---------------------------------------------------------------------------
extern "C" void kernel_launch(void* const* d_in, const int* in_sizes, int n_in,
                              void* d_out, int out_size, void* d_ws, size_t ws_size,
                              hipStream_t stream)
{
    const float* x    = (const float*)d_in[0];
    const float* A    = (const float*)d_in[1];
    const int*   bat  = (const int*)  d_in[2];
    const float* w11 = (const float*)d_in[3],  *b11 = (const float*)d_in[4];
    const float* w12 = (const float*)d_in[5],  *b12 = (const float*)d_in[6];
    const float* w13 = (const float*)d_in[7],  *b13 = (const float*)d_in[8];
    const float* w21 = (const float*)d_in[9],  *b21 = (const float*)d_in[10];
    const float* w22 = (const float*)d_in[11], *b22 = (const float*)d_in[12];
    const float* w23 = (const float*)d_in[13], *b23 = (const float*)d_in[14];
    const float* w31 = (const float*)d_in[15], *b31 = (const float*)d_in[16];
    const float* w32 = (const float*)d_in[17], *b32 = (const float*)d_in[18];
    const float* w33 = (const float*)d_in[19], *b33 = (const float*)d_in[20];

    // workspace layout (floats)
    float* W   = (float*)d_ws;
    float* ha  = W;                          // N*HID (also Ax)
    float* hb  = W + 1048576;                // N*HID (also Axx)
    float* hc  = W + 2 * 1048576;            // N*HID
    float* A3  = W + 3 * 1048576;            // N*N
    float* v   = A3 + (size_t)Nn * Nn;
    float* mseg = v + Nn;
    float* zseg = mseg + NG;
    float* ent  = zseg + NG;
    float* t1   = ent + Nn;
    float* probs = t1 + Nn;
    int* deg    = (int*)(probs + Nn);
    int* rowptr = deg + Nn;
    int* colidx = rowptr + (Nn + 4);

    // output layout
    float* out      = (float*)d_out;
    float* out_hp   = out;                                   // N*D
    float* out_adj  = out + (size_t)Nn * Dd;                 // N*N (A2 staged here)
    float* out_loss = out_adj + (size_t)Nn * Nn;             // 1
    float* out_mask = out_loss + 1;                          // N

    dim3 blk(256);
    dim3 gA(Dd / BN, Nn / BM);     // N=512 GEMMs   (16, 32)
    dim3 gS(Nn / BN, Nn / BM);     // N=2048 GEMMs  (64, 32)

    // CSR extraction
    k_deg <<<Nn, blk, 0, stream>>>(A, deg);
    k_scan<<<1, 32, 0, stream>>>(deg, rowptr);
    k_fill<<<Nn, blk, 0, stream>>>(A, rowptr, colidx);

    // V / entropy pipeline
    gemm_f32_wmma<<<gA, blk, 0, stream>>>(A, x, nullptr, nullptr, ha, Nn, Dd, Nn, 0, 0); // Ax
    gemm_f32_wmma<<<gA, blk, 0, stream>>>(A, x, nullptr, nullptr, hb, Nn, Dd, Nn, 0, 1); // A@(x*x)
    k_vnorm <<<Nn, blk, 0, stream>>>(x, ha, hb, v);
    k_segmax<<<NG, blk, 0, stream>>>(v, bat, mseg);
    k_segz  <<<NG, blk, 0, stream>>>(v, bat, mseg, zseg);
    k_ent   <<<Nn / 256, blk, 0, stream>>>(v, bat, mseg, zseg, ent);
    k_t1    <<<Nn / 256, blk, 0, stream>>>(ent, rowptr, colidx, t1);

    // GNN MLP stack
    k_layer1<<<(Nn * HIDD) / 256, blk, 0, stream>>>(t1, w11, b11, ha);
    gemm_f32_wmma<<<gA, blk, 0, stream>>>(ha, w12, b12, nullptr, hb, Nn, HIDD, HIDD, 1, 0);
    gemm_f32_wmma<<<gA, blk, 0, stream>>>(hb, w13, b13, nullptr, ha, Nn, HIDD, HIDD, 0, 0); // h1
    gemm_f32_wmma<<<gA, blk, 0, stream>>>(A,  ha, nullptr, ha, hb, Nn, HIDD, Nn, 0, 0);     // h1 + A@h1
    gemm_f32_wmma<<<gA, blk, 0, stream>>>(hb, w21, b21, nullptr, hc, Nn, HIDD, HIDD, 1, 0);
    gemm_f32_wmma<<<gA, blk, 0, stream>>>(hc, w22, b22, nullptr, hb, Nn, HIDD, HIDD, 1, 0);
    gemm_f32_wmma<<<gA, blk, 0, stream>>>(hb, w23, b23, nullptr, hc, Nn, HIDD, HIDD, 0, 0); // h2
    gemm_f32_wmma<<<gA, blk, 0, stream>>>(A,  hc, nullptr, hc, hb, Nn, HIDD, Nn, 0, 0);     // h2 + A@h2
    gemm_f32_wmma<<<gA, blk, 0, stream>>>(hb, w31, b31, nullptr, ha, Nn, HIDD, HIDD, 1, 0);
    gemm_f32_wmma<<<gA, blk, 0, stream>>>(ha, w32, b32, nullptr, hb, Nn, HIDD, HIDD, 1, 0);
    k_h3<<<Nn, blk, 0, stream>>>(hb, w33, b33, probs);

    // A^2 (staged in output adj region) and A^3
    gemm_f32_wmma<<<gS, blk, 0, stream>>>(A, A, nullptr, nullptr, out_adj, Nn, Nn, Nn, 0, 0);
    gemm_f32_wmma<<<gS, blk, 0, stream>>>(out_adj, A, nullptr, nullptr, A3, Nn, Nn, Nn, 0, 0);

    // sequential greedy selection -> loss + mask
    k_select<<<1, 1024, 0, stream>>>(ent, probs, rowptr, colidx, out_loss, out_mask);

    // final outputs
    k_hpooled<<<(Nn * Dd) / 256, blk, 0, stream>>>(x, out_mask, out_hp);
    k_adjp   <<<((size_t)Nn * Nn) / 256, blk, 0, stream>>>(out_adj, A3, out_mask);
}